// CropDrones_90271622627839
// MI455X (gfx1250) — compile-verified
//
#include <hip/hip_runtime.h>
#include <hip/hip_bf16.h>
#include <cstdint>

#define B_  32
#define C_  3
#define H_  512
#define W_  512
#define S_  256   // MAX_SIDE
#define SCAN_PARTS 16   // blocks per batch for the mask scan

// CDNA5 async-to-LDS path: probe via __has_builtin so compile never breaks.
#if defined(__HIP_DEVICE_COMPILE__) && defined(__gfx1250__) && \
    __has_builtin(__builtin_amdgcn_global_load_async_to_lds_b128) && \
    __has_builtin(__builtin_amdgcn_s_wait_asynccnt)
#define USE_ASYNC_LDS 1
#else
#define USE_ASYNC_LDS 0
#endif

#if USE_ASYNC_LDS
typedef int v4i_ __attribute__((ext_vector_type(4)));
typedef __attribute__((address_space(1))) v4i_* gptr_v4i;   // global
typedef __attribute__((address_space(3))) v4i_* lptr_v4i;   // LDS

__device__ __forceinline__ void async_copy_b128(const float4* g, float4* l) {
    __builtin_amdgcn_global_load_async_to_lds_b128(
        (gptr_v4i)g, (lptr_v4i)l, /*offset=*/0, /*cpol=*/0);
}
#endif

// ---------------------------------------------------------------- kernel 0
__global__ void bbox_init_kernel(int* __restrict__ bbox) {
    int i = threadIdx.x;
    if (i < B_ * 4) bbox[i] = (i & 1) ? -1 : 0x7fffffff; // [rmin,rmax,cmin,cmax]
}

// ---------------------------------------------------------------- kernel 1
__global__ __launch_bounds__(256)
void bbox_scan_kernel(const float* __restrict__ in1, int* __restrict__ bbox) {
    const int b    = blockIdx.x / SCAN_PARTS;
    const int part = blockIdx.x % SCAN_PARTS;
    const int tid  = threadIdx.x;

    // window mask is channel 3 of (B,4,H,W)
    const float4* win4 =
        (const float4*)(in1 + ((size_t)b * (C_ + 1) + C_) * (size_t)(H_ * W_));

    const int total4    = (H_ * W_) / 4;          // 65536 float4 per batch
    const int per_block = total4 / SCAN_PARTS;    // 4096
    const int base4     = part * per_block;
    const int iters     = per_block / 256;        // 16

    int rmin = 0x7fffffff, rmax = -1, cmin = 0x7fffffff, cmax = -1;

#if USE_ASYNC_LDS
    __shared__ float4 sbuf[2][256];               // 8 KiB double buffer
    // prime the pipeline
    async_copy_b128(win4 + base4 + tid, &sbuf[0][tid]);
    for (int t = 0; t < iters; ++t) {
        if (t + 1 < iters) {
            async_copy_b128(win4 + base4 + (t + 1) * 256 + tid,
                            &sbuf[(t + 1) & 1][tid]);
            __builtin_amdgcn_s_wait_asynccnt(1);  // oldest (tile t) has landed
        } else {
            __builtin_amdgcn_s_wait_asynccnt(0);
        }
        float4 v = sbuf[t & 1][tid];              // each lane reads only its own slot
#else
    for (int t = 0; t < iters; ++t) {
        float4 v = win4[base4 + t * 256 + tid];
#endif
        const int flat = (base4 + t * 256 + tid) * 4;
        const int row  = flat >> 9;               // W_ = 512
        const int col  = flat & (W_ - 1);
        const bool a0 = (v.x != 0.0f), a1 = (v.y != 0.0f),
                   a2 = (v.z != 0.0f), a3 = (v.w != 0.0f);
        if (a0 | a1 | a2 | a3) { rmin = min(rmin, row); rmax = max(rmax, row); }
        if (a0) { cmin = min(cmin, col    ); cmax = max(cmax, col    ); }
        if (a1) { cmin = min(cmin, col + 1); cmax = max(cmax, col + 1); }
        if (a2) { cmin = min(cmin, col + 2); cmax = max(cmax, col + 2); }
        if (a3) { cmin = min(cmin, col + 3); cmax = max(cmax, col + 3); }
    }

    __shared__ int sred[4];
    if (tid == 0) { sred[0] = 0x7fffffff; sred[1] = -1; sred[2] = 0x7fffffff; sred[3] = -1; }
    __syncthreads();
    atomicMin(&sred[0], rmin); atomicMax(&sred[1], rmax);
    atomicMin(&sred[2], cmin); atomicMax(&sred[3], cmax);
    __syncthreads();
    if (tid == 0) {
        atomicMin(&bbox[b * 4 + 0], sred[0]);
        atomicMax(&bbox[b * 4 + 1], sred[1]);
        atomicMin(&bbox[b * 4 + 2], sred[2]);
        atomicMax(&bbox[b * 4 + 3], sred[3]);
    }
}

// ---------------------------------------------------------------- kernel 2
__global__ __launch_bounds__(256)
void crop_kernel(const float* __restrict__ in1, const int* __restrict__ bbox,
                 float* __restrict__ out) {
    const int rowblk = blockIdx.x & 63;           // 64 row-blocks per (b,c)
    const int bc     = blockIdx.x >> 6;
    const int b      = bc / C_;
    const int c      = bc % C_;
    const int y      = rowblk * 4 + threadIdx.y;  // output row
    const int x0     = threadIdx.x * 4;           // 4 output pixels per thread

    const int r0 = bbox[b * 4 + 0];
    const int r1 = bbox[b * 4 + 1];
    const int c0 = bbox[b * 4 + 2];
    const int c1 = bbox[b * 4 + 3];
    const int sides_h = r1 - r0;
    const int sides_w = c1 - c0;
    const int top  = (S_ - sides_h) >> 1;         // positive, floor-div ok
    const int left = (S_ - sides_w) >> 1;

    float4 o = make_float4(0.0f, 0.0f, 0.0f, 0.0f);

    const bool vy = (y >= top) && (y < top + sides_h);
    if (vy) {
        const float* img =
            in1 + ((size_t)b * (C_ + 1) + c) * (size_t)(H_ * W_);
        int sy = y - top + r0;
        sy = min(max(sy, 0), H_ - 1);
        const float* rowp = img + (size_t)sy * W_;
        // gfx1250 prefetch of the segment this thread will gather
        int sx0 = min(max(x0 - left + c0, 0), W_ - 1);
        __builtin_prefetch(rowp + sx0, 0, 0);
        float vals[4];
#pragma unroll
        for (int k = 0; k < 4; ++k) {
            const int x = x0 + k;
            float v = 0.0f;
            if (x >= left && x < left + sides_w) {
                int sx = x - left + c0;
                sx = min(max(sx, 0), W_ - 1);
                v = rowp[sx];
            }
            vals[k] = v;
        }
        o = make_float4(vals[0], vals[1], vals[2], vals[3]);
    }

    float4* outp = (float4*)(out + (((size_t)(b * C_ + c) * S_) + y) * S_ + x0);
    *outp = o;
}

// ---------------------------------------------------------------- launcher
extern "C" void kernel_launch(void* const* d_in, const int* in_sizes, int n_in,
                              void* d_out, int out_size, void* d_ws, size_t ws_size,
                              hipStream_t stream) {
    const float* in1 = (const float*)d_in[0];   // (32,4,512,512) f32
    float*       out = (float*)d_out;           // (32,3,256,256) f32
    int*        bbox = (int*)d_ws;              // 32*4 ints scratch

    bbox_init_kernel<<<1, 128, 0, stream>>>(bbox);
    bbox_scan_kernel<<<B_ * SCAN_PARTS, 256, 0, stream>>>(in1, bbox);
    crop_kernel<<<dim3(B_ * C_ * 64), dim3(64, 4), 0, stream>>>(in1, bbox, out);
}